// NeuroDiversePolicy_8005819040010
// MI455X (gfx1250) — compile-verified
//
#include <hip/hip_runtime.h>
#include <math.h>

typedef __attribute__((ext_vector_type(2))) float v2f;
typedef __attribute__((ext_vector_type(8))) float v8f;

// Types matching the builtin's diagnostic-printed signature:
//   param1: int __attribute__((vector_size(16))) __device__ *   (addrspace(1))
//   param2: (expected) same vector type in __shared__           (addrspace(3))
typedef int v4i __attribute__((vector_size(16)));
typedef __attribute__((address_space(1))) v4i* as1_v4i_p;
typedef __attribute__((address_space(3))) v4i* as3_v4i_p;

#define NN 8192
#define OBS_DIM 64
#define ACTION_DIMS 8
#define KSPLIT 8
#define KSLICE (NN / KSPLIT)            // 1024
#define ROWS_PER_WAVE 16
#define WAVES_PER_BLOCK 4
#define GEMV_BLOCK (WAVES_PER_BLOCK * 32)   // 128 threads (4 wave32)
#define GEMV_GRID ((NN / ROWS_PER_WAVE / WAVES_PER_BLOCK) * KSPLIT)  // 1024 blocks
#define UNROLL 16                        // 16 wmmas (K=64) per loop iter

#if __has_builtin(__builtin_amdgcn_global_load_async_to_lds_b128)
#define HAVE_ASYNC_LDS 1
#else
#define HAVE_ASYNC_LDS 0
#endif

// ---------------------------------------------------------------------------
// Kernel A: x_in = x with obs override; n_active = sum(m)
// ---------------------------------------------------------------------------
__global__ void __launch_bounds__(256)
prep_kernel(const float* __restrict__ obs, const int* __restrict__ m,
            const float* __restrict__ x, float* __restrict__ ws_x,
            int* __restrict__ ws_nactive) {
  __shared__ int red[256];
  const int t = threadIdx.x;
  int acc = 0;
  for (int i = t; i < NN; i += 256) {
    ws_x[i] = (i < OBS_DIM) ? obs[i] : x[i];
    acc += m[i];
  }
  red[t] = acc;
  __syncthreads();
  for (int s = 128; s > 0; s >>= 1) {
    if (t < s) red[t] += red[t + s];
    __syncthreads();
  }
  if (t == 0) *ws_nactive = red[0];
}

// ---------------------------------------------------------------------------
// Kernel B: xp partials via V_WMMA_F32_16X16X4_F32.
//   Each wave: 16 rows of W, one K-slice of 1024.
//   A fragment (16x4 f32, documented layout): lane L<16 holds W[r0+L, k..k+1],
//   lane L>=16 holds W[r0+L-16, k+2..k+3]  -> one global_load_b64 per wmma.
//   B fragment (4x16 f32): every lane in a half-wave feeds the same x values,
//   so ALL 16 columns of B equal x[k..k+3] and every column of D accumulates
//   the same (correct) dot product -- columns of D = A*B + C are independent.
//   D column 0 (lanes 0/16, VGPRs 0..7) is read out for the 16 rows.
// ---------------------------------------------------------------------------
__global__ void __launch_bounds__(GEMV_BLOCK)
gemv_wmma_kernel(const float* __restrict__ w, const float* __restrict__ ws_x,
                 float* __restrict__ ws_part) {
  __shared__ __align__(16) float xs[KSLICE];

  const int slice   = blockIdx.x & (KSPLIT - 1);
  const int rowBlk0 = (blockIdx.x >> 3) * WAVES_PER_BLOCK;
  const int k0      = slice * KSLICE;

  // Stage this block's x slice (4 KB) into LDS.
#if HAVE_ASYNC_LDS
  // CDNA5 async copy: global -> LDS without touching VGPR data; ASYNCcnt.
  // LDS flat addresses carry the byte offset in bits [31:0] (ISA 10.2), so
  // truncating the generic pointer yields the addrspace(3) operand.
#pragma unroll
  for (int j = 0; j < KSLICE; j += GEMV_BLOCK * 4) {
    const int idx = j + threadIdx.x * 4;
    __builtin_amdgcn_global_load_async_to_lds_b128(
        (as1_v4i_p)(uintptr_t)(ws_x + k0 + idx),
        (as3_v4i_p)(unsigned)(uintptr_t)(&xs[idx]),
        0, 0);
  }
  asm volatile("s_wait_asynccnt 0x0" ::: "memory");
#else
  for (int i = threadIdx.x; i < KSLICE; i += GEMV_BLOCK)
    xs[i] = ws_x[k0 + i];
#endif
  __syncthreads();

  const int lane    = threadIdx.x & 31;
  const int wave    = threadIdx.x >> 5;
  const int r0      = (rowBlk0 + wave) * ROWS_PER_WAVE;
  const int row     = r0 + (lane & 15);
  const int colBase = (lane >> 4) * 2;                 // 0 or 2
  const float* wrow = w + (size_t)row * NN + k0 + colBase;

  v8f acc = {};

  for (int k = 0; k < KSLICE; k += UNROLL * 4) {
    // Prefetch the W stream a bit ahead (speculative; safe at tail).
    __builtin_prefetch(wrow + k + UNROLL * 4, 0, 1);

    v2f a[UNROLL];
#pragma unroll
    for (int u = 0; u < UNROLL; ++u)
      a[u] = *(const v2f*)(wrow + k + u * 4);          // global_load_b64

#pragma unroll
    for (int u = 0; u < UNROLL; ++u) {
      const v2f b = *(const v2f*)(&xs[k + u * 4 + colBase]);  // ds_load_b64
      acc = __builtin_amdgcn_wmma_f32_16x16x4_f32(
          /*neg_a=*/false, a[u], /*neg_b=*/false, b,
          /*c_mod=*/(short)0, acc, /*reuse_a=*/false, /*reuse_b=*/false);
    }
  }

  // D column 0: lane 0 has rows r0..r0+7 in acc[0..7], lane 16 rows r0+8..r0+15.
  if ((lane & 15) == 0) {
    float* p = ws_part + slice * NN + r0 + (lane >> 4) * 8;
#pragma unroll
    for (int i = 0; i < 8; ++i) p[i] = acc[i];
  }
}

// ---------------------------------------------------------------------------
// Kernel C: reduce K-split partials, apply per-row 3x2 affine + tanh, scatter.
//   out[0:8]      = a  (y[n_active-8 : n_active])
//   out[8:8+N]    = y
//   out[8+N:8+2N] = h_new
// ---------------------------------------------------------------------------
__global__ void __launch_bounds__(256)
epilogue_kernel(const float* __restrict__ h, const float* __restrict__ nmat,
                const float* __restrict__ ws_part,
                const int* __restrict__ ws_nactive, float* __restrict__ out) {
  const int i = blockIdx.x * 256 + threadIdx.x;
  if (i >= NN) return;

  float xp = 0.0f;
#pragma unroll
  for (int s = 0; s < KSPLIT; ++s) xp += ws_part[s * NN + i];  // fixed order

  const float hv = h[i];
  const float* np = nmat + i * 6;        // n[i, j, o] at j*2 + o
  const float hn = tanhf(hv * np[0] + xp * np[2] + np[4]);
  const float y  = tanhf(hv * np[1] + xp * np[3] + np[5]);

  out[ACTION_DIMS + i]      = y;
  out[ACTION_DIMS + NN + i] = hn;

  int start = *ws_nactive - ACTION_DIMS;
  start = start < 0 ? 0 : (start > NN - ACTION_DIMS ? NN - ACTION_DIMS : start);
  if (i >= start && i < start + ACTION_DIMS) out[i - start] = y;
}

// ---------------------------------------------------------------------------
extern "C" void kernel_launch(void* const* d_in, const int* in_sizes, int n_in,
                              void* d_out, int out_size, void* d_ws, size_t ws_size,
                              hipStream_t stream) {
  const float* obs  = (const float*)d_in[0];
  const float* h    = (const float*)d_in[1];
  const int*   m    = (const int*)d_in[2];
  const float* nmat = (const float*)d_in[3];
  const float* x    = (const float*)d_in[4];
  const float* w    = (const float*)d_in[5];

  float* ws_x     = (float*)d_ws;                 // NN floats
  float* ws_part  = ws_x + NN;                    // KSPLIT * NN floats
  int*   ws_nact  = (int*)(ws_part + KSPLIT * NN);
  float* out      = (float*)d_out;

  prep_kernel<<<1, 256, 0, stream>>>(obs, m, x, ws_x, ws_nact);
  gemv_wmma_kernel<<<GEMV_GRID, GEMV_BLOCK, 0, stream>>>(w, ws_x, ws_part);
  epilogue_kernel<<<(NN + 255) / 256, 256, 0, stream>>>(h, nmat, ws_part,
                                                        ws_nact, out);
}